// Model_7310034338116
// MI455X (gfx1250) — compile-verified
//
#include <hip/hip_runtime.h>
#include <cstddef>

typedef __attribute__((ext_vector_type(2))) float v2f;
typedef __attribute__((ext_vector_type(8))) float v8f;

// D(16x16,f32) = A(16x4,f32) * B(4x16,f32) + C  -- v_wmma_f32_16x16x4_f32
__device__ __forceinline__ v8f wmma4(v2f a, v2f b, v8f c) {
    return __builtin_amdgcn_wmma_f32_16x16x4_f32(
        /*neg_a=*/false, a, /*neg_b=*/false, b,
        /*c_mod=*/(short)0, c, /*reuse_a=*/false, /*reuse_b=*/false);
}

// gfx1250 async global->LDS path (guarded: falls back to VGPR round trip)
#if defined(__has_builtin)
#if __has_builtin(__builtin_amdgcn_global_load_async_to_lds_b128) && \
    __has_builtin(__builtin_amdgcn_s_wait_asynccnt)
#define USE_ASYNC_LDS 1
#endif
#endif
#ifndef USE_ASYNC_LDS
#define USE_ASYNC_LDS 0
#endif

#if USE_ASYNC_LDS
// parameter type per hipcc diagnostic: 'int __attribute__((__vector_size__(16))) *'
typedef int v4i_ld __attribute__((__vector_size__(4 * sizeof(int))));
#endif

// ---------------------------------------------------------------------------
// Zero-padded copy: dst[r,c] = (r<srcRows && c<srcCols) ? src[r,c] : 0
// ---------------------------------------------------------------------------
__global__ __launch_bounds__(256) void pad_copy_kernel(
    const float* __restrict__ src, float* __restrict__ dst, int srcRows,
    int srcCols, int dstCols, int total) {
    const int i = blockIdx.x * 256 + threadIdx.x;
    if (i >= total) return;
    const int r = i / dstCols;
    const int c = i - r * dstCols;
    dst[i] = (r < srcRows && c < srcCols) ? src[(size_t)r * srcCols + c] : 0.0f;
}

// dst[c*rows + r] = src[r*cols + c]
__global__ __launch_bounds__(256) void transpose_kernel(
    const float* __restrict__ src, float* __restrict__ dst, int rows, int cols,
    int total) {
    const int i = blockIdx.x * 256 + threadIdx.x;
    if (i >= total) return;
    const int r = i / cols;
    const int c = i - r * cols;
    dst[(size_t)c * rows + r] = src[i];
}

// ---------------------------------------------------------------------------
// Kernel 1: fused QKV projection on pre-padded inputs (no guards in hot loop).
// ---------------------------------------------------------------------------
template <int HP, int H3, int NH, int DH, bool EXACT>
__global__ __launch_bounds__(32) void qkv_kernel(
    const float* __restrict__ xP, const float* __restrict__ wP,
    const float* __restrict__ qbias, const float* __restrict__ vbias,
    float* __restrict__ Q, float* __restrict__ Kt2, float* __restrict__ V2,
    float rscale) {
    constexpr int L = 1024, DHP = 16;
    const int lane = threadIdx.x;
    const int nn = lane & 15;
    const int g = lane >> 4;
    const int m0 = blockIdx.x * 16;   // token tile
    const int n0 = blockIdx.y * 16;   // output-column tile

    const float* xrow = xP + (size_t)(m0 + nn) * HP + 2 * g;
    const float* wrow = wP + (size_t)(n0 + nn) * HP + 2 * g;

    v8f c = {};
#pragma unroll 4
    for (int kk = 0; kk < HP; kk += 4) {
        const v2f a = *(const v2f*)(xrow + kk);
        const v2f b = *(const v2f*)(wrow + kk);
        c = wmma4(a, b, c);
    }

    const int col = n0 + nn;
    if constexpr (!EXACT) {
        if (col >= H3) return;
    }
    const int hh = col / (3 * DH);
    const int j = col - hh * 3 * DH;
    const int sect = j / DH;
    const int d = j - sect * DH;
#pragma unroll
    for (int r = 0; r < 8; ++r) {
        const int tok = m0 + r + 8 * g;   // C row = r + 8*(lane>>4)
        const int b_ = tok >> 10;
        const int qq = tok & 1023;
        const size_t bh = (size_t)(b_ * NH + hh);
        float val = c[r];
        if (sect == 0) {
            val = (val + qbias[hh * DH + d]) * rscale;
            Q[(bh * L + qq) * DHP + d] = val;
        } else if (sect == 1) {
            Kt2[(bh * 8 + (d >> 1)) * (2 * L) + 2 * qq + (d & 1)] = val;
        } else {
            val += vbias[hh * DH + d];
            V2[(bh * (L / 2) + (qq >> 1)) * 32 + 2 * d + (qq & 1)] = val;
        }
    }
}

// ---------------------------------------------------------------------------
// Kernel 2: fused flash attention, one wave per 16-query tile per (b,h).
// S^T = K_tile * Q^T (softmax rows register-local); O^T = V^T * P^T.
// ---------------------------------------------------------------------------
template <int NH>
__global__ __launch_bounds__(32) void attn_kernel(
    const float* __restrict__ Q, const float* __restrict__ Kt2,
    const float* __restrict__ V2, const int* __restrict__ mask,
    float* __restrict__ CtxT) {
    constexpr int L = 1024, DHP = 16, NT = L / 16, SM = 20;  // LDS row stride
    const int lane = threadIdx.x;
    const int n = lane & 15;   // q index within tile (B/C col)
    const int g = lane >> 4;

    const int qt = blockIdx.x % NT;
    const int bh = blockIdx.x / NT;
    const int h = bh % NH;
    const int b = bh / NH;
    const int q0 = qt * 16;

    const float* Qp = Q + (size_t)(b * NH + h) * L * DHP;
    const float* Ktp = Kt2 + (size_t)(b * NH + h) * DHP * L;   // [8][2L]
    const float* Vp = V2 + (size_t)(b * NH + h) * L * DHP;     // [L/2][32]
    const int* Mp = mask + (size_t)b * L * L;

    __shared__ int sm_mask[16 * SM];
    __shared__ float sm_p[16 * SM];

    // Q^T B-fragments (scale folded in); contiguous pair -> b64 load
    v2f qb[4];
#pragma unroll
    for (int kc = 0; kc < 4; ++kc)
        qb[kc] = *(const v2f*)&Qp[(size_t)(q0 + n) * DHP + 4 * kc + 2 * g];

    v8f o = {};                      // O^T accum: (d = r+8g, q = n)
    float row_max = -3.0e38f;
    float row_sum = 0.0f;

    for (int kt = 0; kt < NT; ++kt) {
        const int k0 = kt * 16;

        // --- mask tile -> LDS, 16B per lane per op ---
#if USE_ASYNC_LDS
#pragma unroll
        for (int i = 0; i < 2; ++i) {
            const int idx4 = i * 128 + lane * 4;   // 0..255 step 4
            const int qq = idx4 >> 4, kk = idx4 & 15;
            __builtin_amdgcn_global_load_async_to_lds_b128(
                (v4i_ld*)&Mp[(size_t)(q0 + qq) * L + (k0 + kk)],
                (v4i_ld*)&sm_mask[qq * SM + kk], 0, 0);
        }
        __builtin_amdgcn_s_wait_asynccnt(0);
#else
#pragma unroll
        for (int i = 0; i < 2; ++i) {
            const int idx4 = i * 128 + lane * 4;   // 0..255 step 4
            const int qq = idx4 >> 4, kk = idx4 & 15;
            *(int4*)&sm_mask[qq * SM + kk] =
                *(const int4*)&Mp[(size_t)(q0 + qq) * L + (k0 + kk)];
        }
#endif
        __syncthreads();

        // --- S^T(16key x 16q) = K_tile(16 x 16d) * Q^T(16d x 16q) ---
        v8f s = {};
#pragma unroll
        for (int kc = 0; kc < 4; ++kc) {
            const v2f a =
                *(const v2f*)&Ktp[(size_t)(2 * kc + g) * (2 * L) + 2 * (k0 + n)];
            s = wmma4(a, qb[kc], s);
        }

        if (kt + 1 < NT) {
            __builtin_prefetch(&Ktp[(size_t)g * (2 * L) + 2 * (k0 + 16 + n)], 0, 1);
            __builtin_prefetch(&Vp[((size_t)((k0 + 16) >> 1) + g) * 32 + 2 * n], 0, 1);
            __builtin_prefetch(&Mp[(size_t)(q0 + (lane >> 1)) * L + (k0 + 16)], 0, 1);
        }

        // --- mask + online softmax (register-local rows) ---
        float keepf[8];
        float tmax = -3.0e38f;
#pragma unroll
        for (int r = 0; r < 8; ++r) {
            const int keep = sm_mask[n * SM + r + 8 * g];
            const float sv = keep ? s[r] : -3.0e38f;
            s[r] = sv;
            keepf[r] = keep ? 1.0f : 0.0f;
            tmax = fmaxf(tmax, sv);
        }
        tmax = fmaxf(tmax, __shfl_xor(tmax, 16, 32));  // merge key halves
        const float new_max = fmaxf(row_max, tmax);
        const float corr = __expf(row_max - new_max);
        float p[8];
        float tsum = 0.0f;
#pragma unroll
        for (int r = 0; r < 8; ++r) {
            const float e = keepf[r] * __expf(s[r] - new_max);
            p[r] = e;
            tsum += e;
        }
        tsum += __shfl_xor(tsum, 16, 32);
        row_sum = row_sum * corr + tsum;
        row_max = new_max;
#pragma unroll
        for (int r = 0; r < 8; ++r) o[r] *= corr;

        __syncthreads();
#pragma unroll
        for (int r = 0; r < 8; ++r) sm_p[(r + 8 * g) * SM + n] = p[r];
        __syncthreads();

        // --- O^T += V^T(16d x 16key) * P^T(16key x 16q) ---
#pragma unroll
        for (int kc = 0; kc < 4; ++kc) {
            const int kk = 4 * kc + 2 * g;   // even key-pair base
            const v2f a =
                *(const v2f*)&Vp[((size_t)((k0 + kk) >> 1)) * 32 + 2 * n];
            v2f bb;
            bb.x = sm_p[kk * SM + n];
            bb.y = sm_p[(kk + 1) * SM + n];
            o = wmma4(a, bb, o);
        }
        __syncthreads();
    }

    // epilogue: normalize, write CtxT[b,h,d,q] (coalesced along q)
    const float inv = 1.0f / fmaxf(row_sum, 1e-20f);
    float* Cp = CtxT + (size_t)(b * NH + h) * DHP * L;
#pragma unroll
    for (int r = 0; r < 8; ++r)
        Cp[(size_t)(r + 8 * g) * L + (q0 + n)] = o[r] * inv;
}

// ---------------------------------------------------------------------------
// Kernel 3: out = LayerNorm(ctx @ w_o^T + b_o + x), one block per token.
// w_oT is the pre-transposed weight: w_oT[c*H + t] = w_o[t*H + c] (coalesced).
// ---------------------------------------------------------------------------
template <int H, int NH, int DH>
__global__ __launch_bounds__(256) void proj_ln_kernel(
    const float* __restrict__ CtxT, const float* __restrict__ x,
    const float* __restrict__ w_oT, const float* __restrict__ b_o,
    const float* __restrict__ ln_g, const float* __restrict__ ln_b,
    float* __restrict__ out) {
    constexpr int L = 1024, DHP = 16;
    const int tok = blockIdx.x;
    const int b_ = tok >> 10;
    const int qq = tok & 1023;
    const int t = threadIdx.x;

    __shared__ float s_ctx[H];
    __shared__ float s_h[H];
    __shared__ float s_red[2][8];

    for (int cI = t; cI < H; cI += 256) {
        const int hh = cI / DH, d = cI % DH;
        s_ctx[cI] = CtxT[((size_t)(b_ * NH + hh) * DHP + d) * L + qq];
    }
    __syncthreads();

    float acc = 0.0f;
    if (t < H) {
        acc = b_o[t] + x[(size_t)tok * H + t];
        for (int cI = 0; cI < H; ++cI)
            acc = fmaf(s_ctx[cI], w_oT[(size_t)cI * H + t], acc);
        s_h[t] = acc;
    }
    __syncthreads();

    float ps = 0.0f, pq = 0.0f;
    for (int iI = t; iI < H; iI += 256) {
        const float v = s_h[iI];
        ps += v;
        pq += v * v;
    }
#pragma unroll
    for (int off = 16; off > 0; off >>= 1) {
        ps += __shfl_xor(ps, off, 32);
        pq += __shfl_xor(pq, off, 32);
    }
    const int wid = t >> 5, lid = t & 31;
    if (lid == 0) { s_red[0][wid] = ps; s_red[1][wid] = pq; }
    __syncthreads();
    if (t == 0) {
        float S = 0.0f, Q2 = 0.0f;
#pragma unroll
        for (int iW = 0; iW < 8; ++iW) { S += s_red[0][iW]; Q2 += s_red[1][iW]; }
        s_red[0][0] = S;
        s_red[1][0] = Q2;
    }
    __syncthreads();
    const float mu = s_red[0][0] / (float)H;
    const float var = s_red[1][0] / (float)H - mu * mu;
    if (t < H) {
        out[(size_t)tok * H + t] =
            (s_h[t] - mu) * rsqrtf(var + 1e-7f) * ln_g[t] + ln_b[t];
    }
}

// ---------------------------------------------------------------------------
extern "C" void kernel_launch(void* const* d_in, const int* in_sizes, int n_in,
                              void* d_out, int out_size, void* d_ws,
                              size_t ws_size, hipStream_t stream) {
    (void)in_sizes; (void)n_in; (void)out_size; (void)ws_size;
    const float* x = (const float*)d_in[0];
    const float* y = (const float*)d_in[1];
    const int* mask0 = (const int*)d_in[2];
    const int* mask1 = (const int*)d_in[3];
    const float* wq0 = (const float*)d_in[4];
    const float* qb0 = (const float*)d_in[5];
    const float* vb0 = (const float*)d_in[6];
    const float* wo0 = (const float*)d_in[7];
    const float* bo0 = (const float*)d_in[8];
    const float* lg0 = (const float*)d_in[9];
    const float* lb0 = (const float*)d_in[10];
    const float* wq1 = (const float*)d_in[11];
    const float* qb1 = (const float*)d_in[12];
    const float* vb1 = (const float*)d_in[13];
    const float* wo1 = (const float*)d_in[14];
    const float* bo1 = (const float*)d_in[15];
    const float* lg1 = (const float*)d_in[16];
    const float* lb1 = (const float*)d_in[17];

    float* out0 = (float*)d_out;
    float* out1 = out0 + (size_t)4 * 1024 * 192;

    float* ws = (float*)d_ws;
    const size_t T0 = (size_t)4 * 12 * 1024 * 16;   // per-tensor floats, cfg0
    const size_t T1 = (size_t)4 * 11 * 1024 * 16;   // per-tensor floats, cfg1
    float* Q0 = ws;
    float* Kt0 = Q0 + T0;
    float* V0 = Kt0 + T0;
    float* C0 = V0 + T0;
    float* Q1 = C0 + T0;
    float* Kt1 = Q1 + T1;
    float* V1 = Kt1 + T1;
    float* C1 = V1 + T1;
    float* xP1 = C1 + T1;                       // [4096, 68] padded y
    float* wP1 = xP1 + (size_t)4096 * 68;       // [208, 68] padded w_qkv1
    float* woT0 = wP1 + (size_t)208 * 68;       // [192,192] transposed w_o0
    float* woT1 = woT0 + (size_t)192 * 192;     // [66,66]   transposed w_o1

    // zero cfg1 Q/Kt/V so the dh=6 -> 16 padding lanes contribute 0
    (void)hipMemsetAsync(Q1, 0, 3 * T1 * sizeof(float), stream);

    // pre-pad cfg1 inputs so the GEMM hot loop needs no bounds checks
    {
        const int totX = 4096 * 68;
        pad_copy_kernel<<<(totX + 255) / 256, 256, 0, stream>>>(y, xP1, 4096,
                                                                66, 68, totX);
        const int totW = 208 * 68;
        pad_copy_kernel<<<(totW + 255) / 256, 256, 0, stream>>>(wq1, wP1, 198,
                                                                66, 68, totW);
        transpose_kernel<<<(192 * 192 + 255) / 256, 256, 0, stream>>>(
            wo0, woT0, 192, 192, 192 * 192);
        transpose_kernel<<<(66 * 66 + 255) / 256, 256, 0, stream>>>(
            wo1, woT1, 66, 66, 66 * 66);
    }

    dim3 w32(32);
    qkv_kernel<192, 576, 12, 16, true><<<dim3(256, 36), w32, 0, stream>>>(
        x, wq0, qb0, vb0, Q0, Kt0, V0, 0.25f);                 // 1/sqrt(16)
    qkv_kernel<68, 198, 11, 6, false><<<dim3(256, 13), w32, 0, stream>>>(
        xP1, wP1, qb1, vb1, Q1, Kt1, V1, 0.40824829046386f);   // 1/sqrt(6)

    attn_kernel<12><<<4 * 12 * 64, w32, 0, stream>>>(Q0, Kt0, V0, mask0, C0);
    attn_kernel<11><<<4 * 11 * 64, w32, 0, stream>>>(Q1, Kt1, V1, mask1, C1);

    proj_ln_kernel<192, 12, 16><<<4096, 256, 0, stream>>>(
        C0, x, woT0, bo0, lg0, lb0, out0);
    proj_ln_kernel<66, 11, 6><<<4096, 256, 0, stream>>>(
        C1, y, woT1, bo1, lg1, lb1, out1);
}